// SECONDBackbone_77902116815211
// MI455X (gfx1250) — compile-verified
//
#include <hip/hip_runtime.h>

typedef __attribute__((ext_vector_type(2))) float v2f;
typedef __attribute__((ext_vector_type(8))) float v8f;

// ---------------------------------------------------------------------------
// Implicit-GEMM 3x3x3 conv, fp32 WMMA 16x16x4, optional fused BN+ReLU+mask
// applied to the *input* gather (params = finalized scale/shift of the
// producing layer, mo = producer's occupancy mask).
//   xin : [B, Di, Hi, Wi, CIN]  channels-last (raw conv out if FUSE, else
//                               pre-masked/zeroed input)
//   w2  : [K, COUT] with K = 27*CIN, k = ((kd*3+kh)*3+kw)*CIN + ci
//   yout: [B, Do, Ho, Wo, COUT] channels-last raw conv output
// One wave = one 16-voxel M-tile x all COUT; 8 waves/block.
// Weight K-chunks (KC = max(CIN,32) rows -> <=32KB) staged through LDS.
// ---------------------------------------------------------------------------
template<int CIN, int NTILES, bool FUSE>
__global__ __launch_bounds__(256)
void conv_wmma(const float* __restrict__ xin, const float* __restrict__ w2,
               const float* __restrict__ params, const float* __restrict__ mo,
               float* __restrict__ yout,
               int B, int Di, int Hi, int Wi,
               int Do, int Ho, int Wo, int stride)
{
  constexpr int COUT = NTILES * 16;
  constexpr int KC   = (CIN < 32) ? 32 : CIN;          // rows per LDS chunk
  constexpr int CSH  = (CIN == 4) ? 2 : (CIN == 16) ? 4 : (CIN == 32) ? 5 : 6;
  constexpr int K    = 27 * CIN;

  __shared__ float wsm[KC * COUT];
  __shared__ float psm[FUSE ? 2 * CIN : 4];

  const int lane  = threadIdx.x & 31;
  const int wave  = threadIdx.x >> 5;
  const int mtile = blockIdx.x * 8 + wave;
  const int Mtot  = B * Do * Ho * Wo;

  if (FUSE) {
    for (int i = threadIdx.x; i < 2 * CIN; i += 256) psm[i] = params[i];
  }

  // A-fragment row for this lane (both half-waves cover M=0..15)
  int  m      = mtile * 16 + (lane & 15);
  bool mvalid = m < Mtot;
  int  mm     = mvalid ? m : 0;
  int xo = mm % Wo; int t = mm / Wo;
  int yo = t % Ho;  t /= Ho;
  int zo = t % Do;  int b = t / Do;
  int zi0 = zo * stride - 1, yi0 = yo * stride - 1, xi0 = xo * stride - 1;
  const int khalf = (lane >> 4) * 2;   // lanes 16-31 hold K offsets {2,3}

  v8f acc[NTILES];
  v8f zero = {0.f,0.f,0.f,0.f,0.f,0.f,0.f,0.f};
#pragma unroll
  for (int nt = 0; nt < NTILES; ++nt) acc[nt] = zero;

  for (int kc = 0; kc < K; kc += KC) {
    int klen = (K - kc < KC) ? (K - kc) : KC;     // always multiple of CIN
    __syncthreads();
    for (int i = threadIdx.x * 4; i < klen * COUT; i += 1024)
      *(float4*)(wsm + i) = *(const float4*)(w2 + kc * COUT + i);
    __syncthreads();
    if (kc + KC < K)
      __builtin_prefetch(w2 + (kc + KC) * COUT, 0, 1);   // next weight chunk

    int ntaps  = klen >> CSH;
    int spbase = kc >> CSH;
    for (int tl = 0; tl < ntaps; ++tl) {
      // decode the 3x3x3 tap once, hoisted over all input channels
      int sp = spbase + tl;
      int kw = sp % 3; int tt = sp / 3; int kh = tt % 3; int kd = tt / 3;
      int zi = zi0 + kd, yi = yi0 + kh, xi = xi0 + kw;
      bool ok = mvalid && (unsigned)zi < (unsigned)Di &&
                (unsigned)yi < (unsigned)Hi && (unsigned)xi < (unsigned)Wi;
      int vox = ((b * Di + zi) * Hi + yi) * Wi + xi;
      const float* xp = xin + (size_t)vox * CIN;
      float movx = 0.f;
      if (FUSE) { if (ok) movx = mo[vox]; }

      int krowbase = tl * CIN * COUT + (lane & 15);
#pragma unroll
      for (int c4 = 0; c4 < CIN; c4 += 4) {
        int ci = c4 + khalf;
        v2f a = {0.f, 0.f};
        if (ok) a = *(const v2f*)(xp + ci);     // one b64 load: same voxel, ci/ci+1
        if (FUSE) {
          float ax = a.x * psm[ci]     + psm[CIN + ci];
          float ay = a.y * psm[ci + 1] + psm[CIN + ci + 1];
          ax = ax > 0.f ? ax : 0.f;
          ay = ay > 0.f ? ay : 0.f;
          a.x = ax * movx;
          a.y = ay * movx;
        }
        int krow = krowbase + ci * COUT;
#pragma unroll
        for (int nt = 0; nt < NTILES; ++nt) {
          v2f bf;
          bf.x = wsm[krow + nt * 16];           // B[k][n]
          bf.y = wsm[krow + COUT + nt * 16];    // B[k+1][n]
          acc[nt] = __builtin_amdgcn_wmma_f32_16x16x4_f32(
              false, a, false, bf, (short)0, acc[nt], false, false);
        }
      }
    }
  }

  // D layout: lane L, VGPR r -> M = r + 8*(L>=16), N = L&15
  int mbase = mtile * 16 + 8 * (lane >> 4);
#pragma unroll
  for (int r = 0; r < 8; ++r) {
    int ms = mbase + r;
    if (ms < Mtot) {
      int xs = ms % Wo; int ts = ms / Wo;
      int ys = ts % Ho; ts /= Ho;
      int zs = ts % Do; int bs = ts / Do;
      int sidx = (((bs * Do + zs) * Ho + ys) * Wo + xs) * COUT + (lane & 15);
#pragma unroll
      for (int nt = 0; nt < NTILES; ++nt)
        yout[sidx + nt * 16] = acc[nt][r];      // lanes -> consecutive channels
    }
  }
}

// ---------------------------------------------------------------------------
// Helper kernels
// ---------------------------------------------------------------------------
__global__ void cvt_mask_kernel(const int* __restrict__ m, float* __restrict__ mf, int n) {
  int i = blockIdx.x * 256 + threadIdx.x;
  if (i < n) mf[i] = (m[i] != 0) ? 1.f : 0.f;
}

// x0[b, s, c] = feat[b, c, s] * mask   (channels-last masked input, Cin=4)
__global__ void make_x0_kernel(const float* __restrict__ feat, const int* __restrict__ mask,
                               float* __restrict__ x0, int B, int C, int cMask, int cShift, int DHW) {
  int total = B * DHW * C;
  for (int i = blockIdx.x * blockDim.x + threadIdx.x; i < total; i += gridDim.x * blockDim.x) {
    int c = i & cMask; int v = i >> cShift;
    int s = v % DHW;   int b = v / DHW;
    float mv = (mask[b * DHW + s] != 0) ? 1.f : 0.f;
    x0[i] = feat[(b * C + c) * DHW + s] * mv;
  }
}

// occupancy pooling: out = 1 if any voxel in 3x3x3 support (pad 1) active
__global__ void maskpool_kernel(const float* __restrict__ mi, float* __restrict__ mo,
                                int B, int Di, int Hi, int Wi, int Do, int Ho, int Wo, int stride) {
  int total = B * Do * Ho * Wo;
  int idx = blockIdx.x * 256 + threadIdx.x;
  if (idx >= total) return;
  int xo = idx % Wo; int t = idx / Wo;
  int yo = t % Ho;   t /= Ho;
  int zo = t % Do;   int b = t / Do;
  float r = 0.f;
  for (int kd = 0; kd < 3; kd++) {
    int z = zo * stride - 1 + kd; if (z < 0 || z >= Di) continue;
    for (int kh = 0; kh < 3; kh++) {
      int y = yo * stride - 1 + kh; if (y < 0 || y >= Hi) continue;
      for (int kw = 0; kw < 3; kw++) {
        int x = xo * stride - 1 + kw; if (x < 0 || x >= Wi) continue;
        if (mi[((b * Di + z) * Hi + y) * Wi + x] > 0.5f) r = 1.f;
      }
    }
  }
  mo[idx] = r;
}

// w [Cout,Cin,27] (OIDHW flat) -> w2 [K,Cout], k = sp*Cin + ci
__global__ void reorder_w_kernel(const float* __restrict__ w, float* __restrict__ w2,
                                 int Cout, int Cin) {
  int K = 27 * Cin;
  int total = K * Cout;
  int i = blockIdx.x * 256 + threadIdx.x;
  if (i >= total) return;
  int co = i % Cout;
  int k  = i / Cout;
  int ci = k % Cin;
  int sp = k / Cin;
  w2[i] = w[(co * Cin + ci) * 27 + sp];
}

__global__ void zero_kernel(float* p, int n) {
  int i = blockIdx.x * 256 + threadIdx.x;
  if (i < n) p[i] = 0.f;
}

// per-channel masked sum/sumsq; channel of each thread is loop-invariant
// because stride (gridDim*256) is a multiple of C (pow2 <= 128)
__global__ void bn_reduce_kernel(const float* __restrict__ y, const float* __restrict__ mo,
                                 float* __restrict__ stats, int Nvox, int C, int cMask, int cShift) {
  __shared__ float sm[256];
  for (int i = threadIdx.x; i < 2 * C; i += blockDim.x) sm[i] = 0.f;
  __syncthreads();
  int c = (blockIdx.x * blockDim.x + threadIdx.x) & cMask;
  float s = 0.f, s2 = 0.f;
  int total = Nvox << cShift;
  for (int i = blockIdx.x * blockDim.x + threadIdx.x; i < total; i += gridDim.x * blockDim.x) {
    int v = i >> cShift;
    if (mo[v] > 0.5f) {
      float val = y[i];
      s  += val;
      s2 += val * val;
    }
  }
  atomicAdd(&sm[c], s);
  atomicAdd(&sm[C + c], s2);
  __syncthreads();
  for (int i = threadIdx.x; i < 2 * C; i += blockDim.x)
    if (sm[i] != 0.f) atomicAdd(&stats[i], sm[i]);
}

__global__ void reduce_sum_kernel(const float* __restrict__ x, int n, float* __restrict__ out) {
  __shared__ float sm[256];
  float s = 0.f;
  for (int i = blockIdx.x * 256 + threadIdx.x; i < n; i += gridDim.x * 256) s += x[i];
  sm[threadIdx.x] = s;
  __syncthreads();
  for (int off = 128; off > 0; off >>= 1) {
    if (threadIdx.x < off) sm[threadIdx.x] += sm[threadIdx.x + off];
    __syncthreads();
  }
  if (threadIdx.x == 0) atomicAdd(out, sm[0]);
}

__global__ void bn_finalize_kernel(const float* __restrict__ stats, const float* __restrict__ g,
                                   const float* __restrict__ bb, float* __restrict__ params, int C) {
  int c = blockIdx.x * blockDim.x + threadIdx.x;
  if (c >= C) return;
  float cnt = stats[2 * C]; if (cnt < 1.f) cnt = 1.f;
  float mean = stats[c] / cnt;
  float var  = stats[C + c] / cnt - mean * mean;
  if (var < 0.f) var = 0.f;
  float inv = rsqrtf(var + 1e-5f);
  float sc  = g[c] * inv;
  params[c]     = sc;
  params[C + c] = bb[c] - mean * sc;
}

// layer-3 apply: channels-last y -> relu(bn(y))*mo -> NCDHW out
__global__ void bn_apply_nchw_kernel(const float* __restrict__ y, const float* __restrict__ mo,
                                     const float* __restrict__ params, float* __restrict__ out,
                                     int B, int DHW, int C, int cMask, int cShift) {
  int total = B * DHW * C;
  for (int i = blockIdx.x * blockDim.x + threadIdx.x; i < total; i += gridDim.x * blockDim.x) {
    int c = i & cMask; int v = i >> cShift;
    int s = v % DHW;   int b = v / DHW;
    float val = y[i] * params[c] + params[C + c];
    val = val > 0.f ? val : 0.f;
    val = (mo[v] > 0.5f) ? val : 0.f;
    out[(b * C + c) * DHW + s] = val;
  }
}

// per-batch masked global average; x already zero at inactive voxels
__global__ void global_pool_kernel(const float* __restrict__ x, const float* __restrict__ mo,
                                   float* __restrict__ gout, int C, int DHW) {
  int b = blockIdx.x / C, c = blockIdx.x % C;
  __shared__ float sm[256], sc[256];
  float s = 0.f, cn = 0.f;
  for (int i = threadIdx.x; i < DHW; i += 256) {
    s  += x[(b * C + c) * DHW + i];
    cn += mo[b * DHW + i];
  }
  sm[threadIdx.x] = s; sc[threadIdx.x] = cn;
  __syncthreads();
  for (int off = 128; off > 0; off >>= 1) {
    if (threadIdx.x < off) { sm[threadIdx.x] += sm[threadIdx.x + off];
                             sc[threadIdx.x] += sc[threadIdx.x + off]; }
    __syncthreads();
  }
  if (threadIdx.x == 0) {
    float cnt = sc[0] < 1.f ? 1.f : sc[0];
    gout[b * C + c] = sm[0] / cnt;
  }
}

// ---------------------------------------------------------------------------
extern "C" void kernel_launch(void* const* d_in, const int* in_sizes, int n_in,
                              void* d_out, int out_size, void* d_ws, size_t ws_size,
                              hipStream_t stream) {
  (void)in_sizes; (void)n_in; (void)out_size; (void)ws_size;
  const float* feat  = (const float*)d_in[0];
  const int*   maskI = (const int*)d_in[1];
  const float* w[4]  = {(const float*)d_in[2],  (const float*)d_in[5],
                        (const float*)d_in[8],  (const float*)d_in[11]};
  const float* g[4]  = {(const float*)d_in[3],  (const float*)d_in[6],
                        (const float*)d_in[9],  (const float*)d_in[12]};
  const float* bb[4] = {(const float*)d_in[4],  (const float*)d_in[7],
                        (const float*)d_in[10], (const float*)d_in[13]};

  const int B = 2;
  const int DHW0 = 32 * 192 * 192, DHW1 = 16 * 96 * 96, DHW2 = 8 * 48 * 48, DHW3 = 4 * 24 * 24;
  const int NV0 = B * DHW0, NV1 = B * DHW1, NV2 = B * DHW2, NV3 = B * DHW3;

  float* base = (float*)d_ws;
  size_t off = 0;
  auto take = [&](size_t n) { float* p = base + off; off += (n + 63) & ~(size_t)63; return p; };

  float* x0  = take((size_t)NV0 * 4);
  float* mF  = take(NV0);
  float* mo0 = take(NV0);
  float* y0  = take((size_t)NV0 * 16);
  float* mo1 = take(NV1);
  float* y1  = take((size_t)NV1 * 32);
  float* mo2 = take(NV2);
  float* y2  = take((size_t)NV2 * 64);
  float* mo3 = take(NV3);
  float* y3  = take((size_t)NV3 * 128);
  float* w2_0 = take(108 * 16);
  float* w2_1 = take(432 * 32);
  float* w2_2 = take(864 * 64);
  float* w2_3 = take(1728 * 128);
  float* stats  = take(257);
  float* params = take(256);

  // masks + channels-last masked input + weight reorder
  cvt_mask_kernel<<<(NV0 + 255) / 256, 256, 0, stream>>>(maskI, mF, NV0);
  make_x0_kernel<<<2048, 256, 0, stream>>>(feat, maskI, x0, B, 4, 3, 2, DHW0);
  maskpool_kernel<<<(NV0 + 255) / 256, 256, 0, stream>>>(mF,  mo0, B, 32, 192, 192, 32, 192, 192, 1);
  maskpool_kernel<<<(NV1 + 255) / 256, 256, 0, stream>>>(mo0, mo1, B, 32, 192, 192, 16, 96, 96, 2);
  maskpool_kernel<<<(NV2 + 255) / 256, 256, 0, stream>>>(mo1, mo2, B, 16, 96, 96, 8, 48, 48, 2);
  maskpool_kernel<<<(NV3 + 255) / 256, 256, 0, stream>>>(mo2, mo3, B, 8, 48, 48, 4, 24, 24, 2);
  reorder_w_kernel<<<(108 * 16 + 255) / 256, 256, 0, stream>>>(w[0], w2_0, 16, 4);
  reorder_w_kernel<<<(432 * 32 + 255) / 256, 256, 0, stream>>>(w[1], w2_1, 32, 16);
  reorder_w_kernel<<<(864 * 64 + 255) / 256, 256, 0, stream>>>(w[2], w2_2, 64, 32);
  reorder_w_kernel<<<(1728 * 128 + 255) / 256, 256, 0, stream>>>(w[3], w2_3, 128, 64);

  // ---- layer 0: 4 -> 16, stride 1 (input pre-masked, no BN fusion) ----
  conv_wmma<4, 1, false><<<(NV0 + 127) / 128, 256, 0, stream>>>(
      x0, w2_0, nullptr, nullptr, y0, B, 32, 192, 192, 32, 192, 192, 1);
  zero_kernel<<<1, 256, 0, stream>>>(stats, 2 * 16 + 1);
  bn_reduce_kernel<<<2048, 256, 0, stream>>>(y0, mo0, stats, NV0, 16, 15, 4);
  reduce_sum_kernel<<<256, 256, 0, stream>>>(mo0, NV0, stats + 2 * 16);
  bn_finalize_kernel<<<1, 64, 0, stream>>>(stats, g[0], bb[0], params, 16);

  // ---- layer 1: 16 -> 32, stride 2 (BN0+ReLU+mask fused into gather) ----
  conv_wmma<16, 2, true><<<(NV1 + 127) / 128, 256, 0, stream>>>(
      y0, w2_1, params, mo0, y1, B, 32, 192, 192, 16, 96, 96, 2);
  zero_kernel<<<1, 256, 0, stream>>>(stats, 2 * 32 + 1);
  bn_reduce_kernel<<<2048, 256, 0, stream>>>(y1, mo1, stats, NV1, 32, 31, 5);
  reduce_sum_kernel<<<128, 256, 0, stream>>>(mo1, NV1, stats + 2 * 32);
  bn_finalize_kernel<<<1, 64, 0, stream>>>(stats, g[1], bb[1], params, 32);

  // ---- layer 2: 32 -> 64, stride 2 ----
  conv_wmma<32, 4, true><<<(NV2 + 127) / 128, 256, 0, stream>>>(
      y1, w2_2, params, mo1, y2, B, 16, 96, 96, 8, 48, 48, 2);
  zero_kernel<<<1, 256, 0, stream>>>(stats, 2 * 64 + 1);
  bn_reduce_kernel<<<1024, 256, 0, stream>>>(y2, mo2, stats, NV2, 64, 63, 6);
  reduce_sum_kernel<<<64, 256, 0, stream>>>(mo2, NV2, stats + 2 * 64);
  bn_finalize_kernel<<<1, 64, 0, stream>>>(stats, g[2], bb[2], params, 64);

  // ---- layer 3: 64 -> 128, stride 2 ----
  conv_wmma<64, 8, true><<<(NV3 + 127) / 128, 256, 0, stream>>>(
      y2, w2_3, params, mo2, y3, B, 8, 48, 48, 4, 24, 24, 2);
  zero_kernel<<<2, 256, 0, stream>>>(stats, 2 * 128 + 1);
  bn_reduce_kernel<<<512, 256, 0, stream>>>(y3, mo3, stats, NV3, 128, 127, 7);
  reduce_sum_kernel<<<32, 256, 0, stream>>>(mo3, NV3, stats + 2 * 128);
  bn_finalize_kernel<<<1, 128, 0, stream>>>(stats, g[3], bb[3], params, 128);

  float* xout = (float*)d_out;
  bn_apply_nchw_kernel<<<512, 256, 0, stream>>>(y3, mo3, params, xout, B, DHW3, 128, 127, 7);
  global_pool_kernel<<<B * 128, 256, 0, stream>>>(xout, mo3,
      xout + (size_t)B * 128 * DHW3, 128, DHW3);
}